// SlideMean_Norm_5643587027445
// MI455X (gfx1250) — compile-verified
//
#include <hip/hip_runtime.h>

// SlideMean_Norm for (32, 1, 16000, 80) f32, window=300 clamped, axis=2.
// Memory-bound: 328 MB min traffic @ 23.3 TB/s => ~14us floor.
// Strategy: one workgroup stages a 512-step tile + +/-150 halo (~254 KB) into
// the 320 KB LDS via async global->LDS B128 copies (ASYNCcnt path), computes
// the exact prefix-sum-difference of the reference in LDS, and streams results
// out with non-temporal stores.

#define B_    32
#define T_    16000
#define F_    80
#define HALF  150
#define TT    512
#define MAXN  (TT + 2 * HALF)          // 812 rows max per tile (with halo)
#define NSEG  ((MAXN + 31) / 32)       // 26 segments of 32 rows
#define NTHREADS 1024

// Builtin signature (from hipcc diagnostic): first param is a pointer to a
// 4-int vector in the global address space; LDS side is the AS(3) analogue.
typedef int v4i_t __attribute__((ext_vector_type(4)));
typedef __attribute__((address_space(1))) v4i_t glb_v4i;
typedef __attribute__((address_space(3))) v4i_t lds_v4i;

__device__ __forceinline__ void copy16_to_lds(const float* g, float* l) {
#if __has_builtin(__builtin_amdgcn_global_load_async_to_lds_b128)
  // Async copy: 16 bytes global -> LDS per lane, tracked by ASYNCcnt.
  __builtin_amdgcn_global_load_async_to_lds_b128(
      (glb_v4i*)g, (lds_v4i*)l, /*offset=*/0, /*cpol=*/0);
#else
  // Fallback: ordinary vector copy through VGPRs.
  *(float4*)l = *(const float4*)g;
#endif
}

__device__ __forceinline__ void wait_async_copies() {
#if __has_builtin(__builtin_amdgcn_s_wait_asynccnt)
  __builtin_amdgcn_s_wait_asynccnt(0);
#else
  asm volatile("s_wait_asynccnt 0" ::: "memory");
#endif
}

__global__ __launch_bounds__(NTHREADS, 1)
void slide_mean_norm_kernel(const float* __restrict__ x,
                            float* __restrict__ out) {
  // 812*80 floats = 259,840 B  +  26*80 floats = 8,320 B   (~262 KB of 320 KB)
  __shared__ float s_data[MAXN * F_];
  __shared__ float s_seg[NSEG * F_];

  const int chunk = blockIdx.x;          // time chunk
  const int b     = blockIdx.y;          // batch
  const int tid   = threadIdx.x;

  const int t0 = chunk * TT;
  const int t1 = min(t0 + TT, T_);
  const int lo = max(t0 - HALF, 0);
  const int hi = min(t1 + HALF, T_);
  const int n  = hi - lo;                // rows staged in LDS (<= MAXN)

  // ---- Phase 1: stage [lo,hi) x 80 (contiguous in memory) into LDS ----
  const float* gx = x + ((size_t)b * T_ + lo) * F_;
  const int total4 = (n * F_) >> 2;      // n*80 is a multiple of 4
  for (int i = tid; i < total4; i += NTHREADS) {
    copy16_to_lds(gx + 4 * i, &s_data[4 * i]);
  }
  wait_async_copies();                   // this wave's asyncs done
  __syncthreads();                       // all waves' asyncs done

  // ---- Phase 2a: inclusive scan within 32-row segments, per feature ----
  const int nseg = (n + 31) >> 5;
  for (int idx = tid; idx < nseg * F_; idx += NTHREADS) {
    const int f   = idx % F_;            // consecutive lanes -> consecutive f
    const int sg  = idx / F_;
    const int r0  = sg << 5;
    const int r1  = min(r0 + 32, n);
    float acc = 0.0f;
    for (int r = r0; r < r1; ++r) {
      acc += s_data[r * F_ + f];
      s_data[r * F_ + f] = acc;
    }
    s_seg[sg * F_ + f] = acc;
  }
  __syncthreads();

  // ---- Phase 2b: exclusive scan of segment totals (one thread per f) ----
  if (tid < F_) {
    float acc = 0.0f;
    for (int sg = 0; sg < nseg; ++sg) {
      const float v = s_seg[sg * F_ + tid];
      s_seg[sg * F_ + tid] = acc;
      acc += v;
    }
  }
  __syncthreads();

  // ---- Phase 2c: add segment offsets -> full inclusive prefix over [lo,hi) --
  for (int idx = tid; idx < n * F_; idx += NTHREADS) {
    const int f = idx % F_;
    const int r = idx / F_;
    s_data[idx] += s_seg[(r >> 5) * F_ + f];
  }
  __syncthreads();

  // ---- Phase 3: out = x - (csum[e] - csum[s]) / (e - s) ----
  const size_t obase = ((size_t)b * T_ + t0) * F_;
  const float* gxt  = x + obase;         // re-read x (L2-hot), exact values
  float*       gout = out + obase;
  const int nout = (t1 - t0) * F_;
  for (int idx = tid; idx < nout; idx += NTHREADS) {
    const int f  = idx % F_;
    const int tt = idx / F_;
    const int t  = t0 + tt;
    const int s  = max(t - HALF, 0);
    const int e  = min(t + HALF, T_ - 1);   // e > s always (count 150..300)
    const float Ie = s_data[(e - lo - 1) * F_ + f];
    const float Is = (s > lo) ? s_data[(s - lo - 1) * F_ + f] : 0.0f;
    const float xv = gxt[idx];
    const float mean = (Ie - Is) / (float)(e - s);
    __builtin_nontemporal_store(xv - mean, &gout[idx]);
  }
}

extern "C" void kernel_launch(void* const* d_in, const int* in_sizes, int n_in,
                              void* d_out, int out_size, void* d_ws, size_t ws_size,
                              hipStream_t stream) {
  (void)in_sizes; (void)n_in; (void)out_size; (void)d_ws; (void)ws_size;
  const float* x   = (const float*)d_in[0];
  float*       out = (float*)d_out;

  dim3 grid((T_ + TT - 1) / TT, B_);   // 32 time chunks x 32 batches
  slide_mean_norm_kernel<<<grid, NTHREADS, 0, stream>>>(x, out);
}